// Gra_Encoder_8873402434056
// MI455X (gfx1250) — compile-verified
//
#include <hip/hip_runtime.h>

// ---------------------------------------------------------------------------
// GAT-VGAE encoder for MI455X (gfx1250, wave32).
// GEMMs via V_WMMA_F32_16X16X4_F32; edge softmax/scatter via global atomics.
// ---------------------------------------------------------------------------

#define N_NODES 32000
#define N_EDGES 512000
#define E_TOT   (N_EDGES + N_NODES)   // + self loops
#define F_IN    128
#define HEADS   4
#define CH      64
#define HC      256                   // HEADS*CH
#define BATCH   64
#define HID     512
#define ZDIM    64
#define XG_K    (N_NODES / BATCH * HC) // 128000

typedef __attribute__((ext_vector_type(2))) float v2f;
typedef __attribute__((ext_vector_type(4))) float v4f;
typedef __attribute__((ext_vector_type(8))) float v8f;

// ---- f32 WMMA 16x16x4 (CDNA5) ---------------------------------------------
__device__ __forceinline__ v8f wmma_f32_4(v2f a, v2f b, v8f c) {
#if __has_builtin(__builtin_amdgcn_wmma_f32_16x16x4_f32)
  return __builtin_amdgcn_wmma_f32_16x16x4_f32(
      /*neg_a=*/false, a, /*neg_b=*/false, b,
      /*c_mod=*/(short)0, c, /*reuse_a=*/false, /*reuse_b=*/false);
#else
  c[0] += a.x * b.x + a.y * b.y;  // compile-only safety net
  return c;
#endif
}

// Monotone-uint encoding of float for deterministic atomicMax on f32.
__device__ __forceinline__ unsigned fenc(float f) {
  unsigned u = __float_as_uint(f);
  return (u & 0x80000000u) ? ~u : (u | 0x80000000u);
}
__device__ __forceinline__ float fdec(unsigned e) {
  return (e & 0x80000000u) ? __uint_as_float(e ^ 0x80000000u)
                           : __uint_as_float(~e);
}
__device__ __forceinline__ float lrelu02(float v) { return v > 0.f ? v : 0.2f * v; }

// ---------------------------------------------------------------------------
// GEMM: H[N_NODES, 256] = X[N_NODES, K] @ W[K, 256]   (K = 128 or 256)
// grid = (N_NODES/16, 4), block = 128 (4 waves); wave w -> 16-col tile.
// WMMA f32 16x16x4 VGPR layout:
//   A (16x4): lane<16 holds {A[m][k0],A[m][k0+1]}, lane>=16 {A[m][k0+2],A[m][k0+3]}
//   B (4x16): mirrored (vgpr j, half h -> K = k0 + 2*h + j, N = lane&15)
//   C/D: vgpr r -> M = r + 8*(lane>=16), N = lane&15
// ---------------------------------------------------------------------------
template <int K>
__global__ void gemm_node_kernel(const float* __restrict__ X,
                                 const float* __restrict__ W,
                                 float* __restrict__ H) {
  const int wave = threadIdx.x >> 5;
  const int lane = threadIdx.x & 31;
  const int m0 = blockIdx.x * 16;
  const int n0 = blockIdx.y * 64 + wave * 16;
  const int mr = m0 + (lane & 15);
  const int nc = n0 + (lane & 15);
  const int kh = (lane >> 4) * 2;

  v8f acc = {};
#pragma unroll 8
  for (int k = 0; k < K; k += 4) {
    v2f a, b;
    a.x = X[mr * K + k + kh];
    a.y = X[mr * K + k + kh + 1];
    b.x = W[(k + kh) * HC + nc];
    b.y = W[(k + kh + 1) * HC + nc];
    acc = wmma_f32_4(a, b, acc);
  }
  const int mbase = m0 + (lane >> 4) * 8;
#pragma unroll
  for (int r = 0; r < 8; ++r) H[(mbase + r) * HC + nc] = acc[r];
}

// a_s[n,h] = sum_c H[n,h*64+c] * att_src[h,c]; likewise a_d.
__global__ void att_kernel(const float* __restrict__ H,
                           const float* __restrict__ att_src,
                           const float* __restrict__ att_dst,
                           float* __restrict__ a_s, float* __restrict__ a_d) {
  int t = blockIdx.x * blockDim.x + threadIdx.x;
  if (t >= N_NODES * HEADS) return;
  int n = t >> 2, h = t & 3;
  const float* hp = H + n * HC + h * CH;
  const float* as = att_src + h * CH;
  const float* ad = att_dst + h * CH;
  float s = 0.f, d = 0.f;
#pragma unroll 8
  for (int c = 0; c < CH; ++c) { float v = hp[c]; s += v * as[c]; d += v * ad[c]; }
  a_s[t] = s; a_d[t] = d;
}

// Zero agg buffer, max-encodings and denominators.
__global__ void init_layer_kernel(float* __restrict__ agg,
                                  unsigned* __restrict__ menc,
                                  float* __restrict__ denom) {
  int t = blockIdx.x * blockDim.x + threadIdx.x;
  if (t < N_NODES * HC) agg[t] = 0.f;
  if (t < N_NODES * HEADS) { menc[t] = 0u; denom[t] = 0.f; }
}

__device__ __forceinline__ void edge_sd(const int* __restrict__ ei, int e,
                                        int& s, int& d) {
  if (e < N_EDGES) { s = ei[e]; d = ei[N_EDGES + e]; }
  else             { s = d = e - N_EDGES; }   // appended self loops
}

// Pass 1: segment max of leaky_relu(a_s[src]+a_d[dst]) over dst.
__global__ void edge_max_kernel(const int* __restrict__ ei,
                                const float* __restrict__ a_s,
                                const float* __restrict__ a_d,
                                unsigned* __restrict__ menc) {
  int t = blockIdx.x * blockDim.x + threadIdx.x;
  if (t >= E_TOT * HEADS) return;
  int e = t >> 2, h = t & 3, s, d;
  edge_sd(ei, e, s, d);
  float v = lrelu02(a_s[s * 4 + h] + a_d[d * 4 + h]);
  atomicMax(&menc[d * 4 + h], fenc(v));
}

// Pass 2: denom[dst] += exp(e - m[dst]).
__global__ void edge_sum_kernel(const int* __restrict__ ei,
                                const float* __restrict__ a_s,
                                const float* __restrict__ a_d,
                                const unsigned* __restrict__ menc,
                                float* __restrict__ denom) {
  int t = blockIdx.x * blockDim.x + threadIdx.x;
  if (t >= E_TOT * HEADS) return;
  int e = t >> 2, h = t & 3, s, d;
  edge_sd(ei, e, s, d);
  float v = lrelu02(a_s[s * 4 + h] + a_d[d * 4 + h]);
  atomicAdd(&denom[d * 4 + h], __expf(v - fdec(menc[d * 4 + h])));
}

// Pass 3: agg[dst, :] += h[src, :] * alpha(head). One wave per EDGE (all 4
// heads): lane owns 8 contiguous columns -> two global_load_b128 per lane,
// the wave reads the full 1KB h[src] row coalesced. Per-head alpha computed
// on lanes 0..3 and broadcast via wave32 __shfl.
__global__ void edge_agg_kernel(const int* __restrict__ ei,
                                const float* __restrict__ a_s,
                                const float* __restrict__ a_d,
                                const unsigned* __restrict__ menc,
                                const float* __restrict__ denom,
                                const float* __restrict__ H,
                                float* __restrict__ agg) {
  int w = (blockIdx.x * blockDim.x + threadIdx.x) >> 5;   // global wave id = edge
  int lane = threadIdx.x & 31;
  if (w >= E_TOT) return;
  int s, d;
  edge_sd(ei, w, s, d);

  // alpha for head (lane & 3); only lanes 0..3 hold distinct heads.
  int h = lane & 3;
  float v = lrelu02(a_s[s * 4 + h] + a_d[d * 4 + h]);
  float alpha_h = __expf(v - fdec(menc[d * 4 + h])) / denom[d * 4 + h];
  float alpha = __shfl(alpha_h, lane >> 3, 32);   // head of cols lane*8..lane*8+7

  const v4f h0 = *(const v4f*)(H + s * HC + lane * 8);
  const v4f h1 = *(const v4f*)(H + s * HC + lane * 8 + 4);
  float* op = agg + d * HC + lane * 8;
  atomicAdd(op + 0, h0.x * alpha);
  atomicAdd(op + 1, h0.y * alpha);
  atomicAdd(op + 2, h0.z * alpha);
  atomicAdd(op + 3, h0.w * alpha);
  atomicAdd(op + 4, h1.x * alpha);
  atomicAdd(op + 5, h1.y * alpha);
  atomicAdd(op + 6, h1.z * alpha);
  atomicAdd(op + 7, h1.w * alpha);
}

// out = relu(agg + bias)
__global__ void bias_relu_kernel(const float* __restrict__ agg,
                                 const float* __restrict__ bias,
                                 float* __restrict__ out) {
  int t = blockIdx.x * blockDim.x + threadIdx.x;
  if (t >= N_NODES * HC) return;
  float v = agg[t] + bias[t & (HC - 1)];
  out[t] = v > 0.f ? v : 0.f;
}

// Y[b, j] = be[j]  (bias preloaded, GEMM accumulates on top)
__global__ void init_y_kernel(float* __restrict__ Y, const float* __restrict__ be) {
  int t = blockIdx.x * blockDim.x + threadIdx.x;
  if (t < BATCH * HID) Y[t] = be[t & (HID - 1)];
}

// Y += XG[64,128000] @ We[128000,512], K-split across blockIdx.z.
// grid = (4, 32, 125), block = 32 (one wave per 16x16 tile-chunk).
// One prefetch per 128-wide k-block; inner WMMA chain unrolled x8 so the
// loads batch into clauses and stay in flight.
#define KSPLIT 125
#define KCHUNK (XG_K / KSPLIT)   // 1024
__global__ void gemm_enc_kernel(const float* __restrict__ XG,
                                const float* __restrict__ We,
                                float* __restrict__ Y) {
  const int lane = threadIdx.x;
  const int m0 = blockIdx.x * 16, n0 = blockIdx.y * 16;
  const int k0 = blockIdx.z * KCHUNK;
  const int mr = m0 + (lane & 15);
  const int nc = n0 + (lane & 15);
  const int kh = (lane >> 4) * 2;

  v8f acc = {};
  for (int kb = k0; kb < k0 + KCHUNK; kb += 128) {
    if (kb + 256 < k0 + KCHUNK)
      __builtin_prefetch(&We[(kb + 256 + kh) * HID + nc], 0, 1);  // global_prefetch_b8
#pragma unroll 8
    for (int k = kb; k < kb + 128; k += 4) {
      v2f a, b;
      a.x = XG[mr * XG_K + k + kh];
      a.y = XG[mr * XG_K + k + kh + 1];
      b.x = We[(k + kh) * HID + nc];
      b.y = We[(k + kh + 1) * HID + nc];
      acc = wmma_f32_4(a, b, acc);
    }
  }
  const int mbase = m0 + (lane >> 4) * 8;
#pragma unroll
  for (int r = 0; r < 8; ++r) atomicAdd(&Y[(mbase + r) * HID + nc], acc[r]);
}

// BatchNorm over batch dim (training stats, eps=1e-3), in place. 512 columns.
__global__ void bn_kernel(float* __restrict__ Y) {
  int j = blockIdx.x * blockDim.x + threadIdx.x;
  if (j >= HID) return;
  float s = 0.f, s2 = 0.f;
  for (int b = 0; b < BATCH; ++b) { float v = Y[b * HID + j]; s += v; s2 += v * v; }
  float mu = s * (1.f / BATCH);
  float var = s2 * (1.f / BATCH) - mu * mu;
  float inv = rsqrtf(var + 1e-3f);
  for (int b = 0; b < BATCH; ++b) Y[b * HID + j] = (Y[b * HID + j] - mu) * inv;
}

// LayerNorm per row (eps=1e-5) + relu, in place. One block per row.
__global__ void ln_relu_kernel(float* __restrict__ Y) {
  __shared__ float red0[256], red1[256];
  const int b = blockIdx.x, tid = threadIdx.x;
  float s = 0.f, s2 = 0.f;
  for (int j = tid; j < HID; j += 256) { float v = Y[b * HID + j]; s += v; s2 += v * v; }
  red0[tid] = s; red1[tid] = s2;
  __syncthreads();
  for (int off = 128; off > 0; off >>= 1) {
    if (tid < off) { red0[tid] += red0[tid + off]; red1[tid] += red1[tid + off]; }
    __syncthreads();
  }
  float mu = red0[0] * (1.f / HID);
  float var = red1[0] * (1.f / HID) - mu * mu;
  float inv = rsqrtf(var + 1e-5f);
  for (int j = tid; j < HID; j += 256) {
    float v = (Y[b * HID + j] - mu) * inv;
    Y[b * HID + j] = v > 0.f ? v : 0.f;
  }
}

// z_loc = Y@Wmu + bmu ; z_scale = exp(Y@Wvar + bvar) + 1e-4, fused into d_out.
// grid = (4, 4, 2), block = 32.
__global__ void head_kernel(const float* __restrict__ Y,
                            const float* __restrict__ Wmu, const float* __restrict__ bmu,
                            const float* __restrict__ Wvar, const float* __restrict__ bvar,
                            float* __restrict__ out) {
  const int lane = threadIdx.x;
  const int m0 = blockIdx.x * 16, n0 = blockIdx.y * 16;
  const int which = blockIdx.z;   // 0 = z_loc, 1 = z_scale
  const float* W = which ? Wvar : Wmu;
  const float* bz = which ? bvar : bmu;
  const int mr = m0 + (lane & 15);
  const int nc = n0 + (lane & 15);
  const int kh = (lane >> 4) * 2;

  v8f acc = {};
#pragma unroll 8
  for (int k = 0; k < HID; k += 4) {
    v2f a, b;
    a.x = Y[mr * HID + k + kh];
    a.y = Y[mr * HID + k + kh + 1];
    b.x = W[(k + kh) * ZDIM + nc];
    b.y = W[(k + kh + 1) * ZDIM + nc];
    acc = wmma_f32_4(a, b, acc);
  }
  const int mbase = m0 + (lane >> 4) * 8;
#pragma unroll
  for (int r = 0; r < 8; ++r) {
    float v = acc[r] + bz[nc];
    if (which) v = __expf(v) + 1e-4f;
    out[which * (BATCH * ZDIM) + (mbase + r) * ZDIM + nc] = v;
  }
}

// ---------------------------------------------------------------------------
extern "C" void kernel_launch(void* const* d_in, const int* in_sizes, int n_in,
                              void* d_out, int out_size, void* d_ws, size_t ws_size,
                              hipStream_t stream) {
  const float* x        = (const float*)d_in[0];
  const float* W1       = (const float*)d_in[1];
  const float* att_src1 = (const float*)d_in[2];
  const float* att_dst1 = (const float*)d_in[3];
  const float* b1       = (const float*)d_in[4];
  const float* W2       = (const float*)d_in[5];
  const float* att_src2 = (const float*)d_in[6];
  const float* att_dst2 = (const float*)d_in[7];
  const float* b2       = (const float*)d_in[8];
  const float* We       = (const float*)d_in[9];
  const float* be       = (const float*)d_in[10];
  const float* Wmu      = (const float*)d_in[11];
  const float* bmu      = (const float*)d_in[12];
  const float* Wvar     = (const float*)d_in[13];
  const float* bvar     = (const float*)d_in[14];
  const int*   ei       = (const int*)d_in[15];   // edge_index [2, E]
  float* out = (float*)d_out;

  // Workspace layout (floats)
  float* ws = (float*)d_ws;
  float*    bufA  = ws;                       // [32000,256] linear transform h
  float*    bufB  = bufA + N_NODES * HC;      // [32000,256] attention aggregate
  float*    bufC  = bufB + N_NODES * HC;      // [32000,256] layer output
  float*    a_s   = bufC + N_NODES * HC;      // [32000,4]
  float*    a_d   = a_s + N_NODES * HEADS;
  unsigned* menc  = (unsigned*)(a_d + N_NODES * HEADS);
  float*    denom = (float*)menc + N_NODES * HEADS;
  float*    Y     = denom + N_NODES * HEADS;  // [64,512]

  const int NH   = N_NODES * HEADS;       // 128000
  const int EH   = E_TOT * HEADS;         // 2176000
  const int NELT = N_NODES * HC;          // 8192000

  // ---------------- Layer 1 ----------------
  gemm_node_kernel<F_IN><<<dim3(N_NODES / 16, 4), 128, 0, stream>>>(x, W1, bufA);
  att_kernel<<<(NH + 255) / 256, 256, 0, stream>>>(bufA, att_src1, att_dst1, a_s, a_d);
  init_layer_kernel<<<(NELT + 255) / 256, 256, 0, stream>>>(bufB, menc, denom);
  edge_max_kernel<<<(EH + 255) / 256, 256, 0, stream>>>(ei, a_s, a_d, menc);
  edge_sum_kernel<<<(EH + 255) / 256, 256, 0, stream>>>(ei, a_s, a_d, menc, denom);
  edge_agg_kernel<<<(E_TOT * 32 + 255) / 256, 256, 0, stream>>>(ei, a_s, a_d, menc,
                                                                denom, bufA, bufB);
  bias_relu_kernel<<<(NELT + 255) / 256, 256, 0, stream>>>(bufB, b1, bufC);

  // ---------------- Layer 2 ----------------
  gemm_node_kernel<HC><<<dim3(N_NODES / 16, 4), 128, 0, stream>>>(bufC, W2, bufA);
  att_kernel<<<(NH + 255) / 256, 256, 0, stream>>>(bufA, att_src2, att_dst2, a_s, a_d);
  init_layer_kernel<<<(NELT + 255) / 256, 256, 0, stream>>>(bufB, menc, denom);
  edge_max_kernel<<<(EH + 255) / 256, 256, 0, stream>>>(ei, a_s, a_d, menc);
  edge_sum_kernel<<<(EH + 255) / 256, 256, 0, stream>>>(ei, a_s, a_d, menc, denom);
  edge_agg_kernel<<<(E_TOT * 32 + 255) / 256, 256, 0, stream>>>(ei, a_s, a_d, menc,
                                                                denom, bufA, bufB);
  bias_relu_kernel<<<(NELT + 255) / 256, 256, 0, stream>>>(bufB, b2, bufC);

  // ---------------- Encoder head ----------------
  init_y_kernel<<<(BATCH * HID + 255) / 256, 256, 0, stream>>>(Y, be);
  gemm_enc_kernel<<<dim3(BATCH / 16, HID / 16, KSPLIT), 32, 0, stream>>>(bufC, We, Y);
  bn_kernel<<<(HID + 255) / 256, 256, 0, stream>>>(Y);
  ln_relu_kernel<<<BATCH, 256, 0, stream>>>(Y);
  head_kernel<<<dim3(BATCH / 16, ZDIM / 16, 2), 32, 0, stream>>>(Y, Wmu, bmu, Wvar, bvar,
                                                                 out);
}